// FocalLoss_31112743092366
// MI455X (gfx1250) — compile-verified
//
#include <hip/hip_runtime.h>
#include <hip/hip_bf16.h>

#define Bc 8
#define Ac 100000
#define Cc 80
#define Mc 32
#define ALPHA_ 0.25f

typedef __attribute__((ext_vector_type(4))) unsigned int uint4v;
typedef __attribute__((ext_vector_type(8))) int          int8v;
typedef __attribute__((ext_vector_type(4))) int          int4v;
typedef __attribute__((ext_vector_type(4))) float        f4v;

// ---------------- block reduction (wave32) ----------------
__device__ inline float block_reduce_sum(float v, float* sm) {
    // wave32 butterfly
    #pragma unroll
    for (int off = 16; off > 0; off >>= 1) v += __shfl_xor(v, off, 32);
    const int lane = threadIdx.x & 31;
    const int w    = threadIdx.x >> 5;
    if (lane == 0) sm[w] = v;
    __syncthreads();
    if (w == 0) {
        v = (lane < (int)(blockDim.x >> 5)) ? sm[lane] : 0.0f;
        #pragma unroll
        for (int off = 4; off > 0; off >>= 1) v += __shfl_xor(v, off, 32);
    }
    return v; // valid in thread 0
}

// ---------------- Kernel A: anchor assignment + regression loss ----------------
__global__ __launch_bounds__(256) void fl_assign_kernel(
    const float* __restrict__ anchors,      // [A,4]   (read 8x -> keep cached)
    const float* __restrict__ annotations,  // [B,M,5] (staged via TDM)
    const float* __restrict__ regs,         // [B,A,4] (read once -> NT)
    signed char* __restrict__ targ,         // [B,A]  (class / -1 neg / -2 ignore)
    float* __restrict__ acc)                // [0..7]=cls_sum [8..15]=reg_sum [16..23]=num_pos
{
    __shared__ float s_ann[Mc * 5];
    __shared__ float s_r1[8], s_r2[8];
    const int b = blockIdx.y;
    const int a = blockIdx.x * blockDim.x + threadIdx.x;

#if defined(__gfx1250__) && __has_builtin(__builtin_amdgcn_tensor_load_to_lds)
    // Stage annotations[b] (32 rows x 5 dwords) into LDS via the Tensor Data Mover.
    if (threadIdx.x == 0) {
        unsigned long long ga =
            (unsigned long long)(const void*)(annotations + (size_t)b * Mc * 5);
        unsigned int lds = (unsigned int)(unsigned long long)(void*)&s_ann[0];
        uint4v g0;
        g0[0] = 1u;                                           // count=1, user descriptor
        g0[1] = lds;                                          // lds_addr (bytes)
        g0[2] = (unsigned int)(ga & 0xFFFFFFFFu);             // global_addr[31:0]
        g0[3] = (unsigned int)((ga >> 32) & 0x1FFFFFFu)       // global_addr[56:32]
              | (2u << 30);                                   // type = 2 ("image")
        int8v g1;
        g1[0] = (int)(2u << 16);     // data_size = 2 (4 bytes); workgroup_mask = 0
        g1[1] = (int)(5u << 16);     // tensor_dim0[15:0] = 5
        g1[2] = (int)(32u << 16);    // tensor_dim1[15:0] = 32
        g1[3] = (int)(5u << 16);     // tile_dim0 = 5
        g1[4] = 32;                  // tile_dim1 = 32, tile_dim2 = 0
        g1[5] = 5;                   // tensor_dim0_stride = 5
        g1[6] = 0;                   // tensor_dim1_stride (unused, 2D)
        g1[7] = 0;
        int4v z4 = {0, 0, 0, 0};
        int8v z8 = {0, 0, 0, 0, 0, 0, 0, 0};
        __builtin_amdgcn_tensor_load_to_lds(g0, g1, z4, z4, z8, 0);
        __builtin_amdgcn_s_wait_tensorcnt(0);
    }
#else
    if (threadIdx.x < Mc * 5)
        s_ann[threadIdx.x] = annotations[(size_t)b * Mc * 5 + threadIdx.x];
#endif
    __syncthreads();

    float reg_partial = 0.0f;
    float npos_partial = 0.0f;

    if (a < Ac) {
        const f4v an = ((const f4v*)anchors)[a];
        const float ax1 = an.x, ay1 = an.y, ax2 = an.z, ay2 = an.w;
        const float aw = ax2 - ax1;
        const float ah = ay2 - ay1;
        const float area_a = aw * ah;

        float best = -2.0f;
        int   bj   = 0;
        #pragma unroll 4
        for (int j = 0; j < Mc; ++j) {
            const float bx1 = s_ann[j * 5 + 0];
            const float by1 = s_ann[j * 5 + 1];
            const float bx2 = s_ann[j * 5 + 2];
            const float by2 = s_ann[j * 5 + 3];
            const float bcl = s_ann[j * 5 + 4];
            float iw = fminf(ax2, bx2) - fmaxf(ax1, bx1);
            float ih = fminf(ay2, by2) - fmaxf(ay1, by1);
            iw = fmaxf(iw, 0.0f);
            ih = fmaxf(ih, 0.0f);
            const float inter = iw * ih;
            const float ua =
                fmaxf(area_a + (bx2 - bx1) * (by2 - by1) - inter, 1e-8f);
            const float iou = (bcl != -1.0f) ? (inter / ua) : -1.0f;
            if (iou > best) { best = iou; bj = j; }   // first-occurrence argmax
        }
        const bool pos = best >= 0.5f;
        const bool neg = best <  0.4f;
        signed char t = pos ? (signed char)(int)s_ann[bj * 5 + 4]
                            : (neg ? (signed char)(-1) : (signed char)(-2));
        targ[(size_t)b * Ac + a] = t;

        if (pos) {
            const float gx1 = s_ann[bj * 5 + 0];
            const float gy1 = s_ann[bj * 5 + 1];
            const float gx2 = s_ann[bj * 5 + 2];
            const float gy2 = s_ann[bj * 5 + 3];
            const float gw  = fmaxf(gx2 - gx1, 0.0f);
            const float gh  = fmaxf(gy2 - gy1, 0.0f);
            const float gcx = gx1 + 0.5f * (gx2 - gx1);
            const float gcy = gy1 + 0.5f * (gy2 - gy1);
            const float acx = ax1 + 0.5f * aw;
            const float acy = ay1 + 0.5f * ah;
            const f4v rp =
                __builtin_nontemporal_load((const f4v*)regs + (size_t)b * Ac + a);
            // targets / NORM  (NORM = 0.1,0.1,0.2,0.2)
            const float t0 = ((gcx - acx) / aw) * 10.0f;
            const float t1 = ((gcy - acy) / ah) * 10.0f;
            const float t2 = __logf(gw / aw) * 5.0f;
            const float t3 = __logf(gh / ah) * 5.0f;
            const float d0 = fabsf(t0 - rp.x);
            const float d1 = fabsf(t1 - rp.y);
            const float d2 = fabsf(t2 - rp.z);
            const float d3 = fabsf(t3 - rp.w);
            const float k = 1.0f / 9.0f, c = 5.0f / 9.0f;
            reg_partial  = (d0 <= k ? 4.5f * d0 * d0 : d0 - c)
                         + (d1 <= k ? 4.5f * d1 * d1 : d1 - c)
                         + (d2 <= k ? 4.5f * d2 * d2 : d2 - c)
                         + (d3 <= k ? 4.5f * d3 * d3 : d3 - c);
            npos_partial = 1.0f;
        }
    }

    const float rsum = block_reduce_sum(reg_partial, s_r1);
    const float nsum = block_reduce_sum(npos_partial, s_r2);
    if (threadIdx.x == 0) {
        atomicAdd(&acc[Bc + b],     rsum);
        atomicAdd(&acc[2 * Bc + b], nsum);
    }
}

// ---------------- Kernel B: streaming focal classification loss ----------------
__global__ __launch_bounds__(256) void fl_cls_kernel(
    const float* __restrict__ cls,          // [B,A,C]  (256 MB read-once -> NT)
    const signed char* __restrict__ targ,   // [B,A]    (reused 20x -> cached)
    float* __restrict__ acc)
{
    __shared__ float s_red[8];
    const int b = blockIdx.y;
    const int nchunks = Ac * Cc / 4;        // 2,000,000 float4 chunks per image
    const f4v* __restrict__ cp = (const f4v*)(cls + (size_t)b * Ac * Cc);
    const signed char* __restrict__ tp = targ + (size_t)b * Ac;

    float sum = 0.0f;
    const int stride = gridDim.x * blockDim.x;
    for (int i = blockIdx.x * blockDim.x + threadIdx.x; i < nchunks; i += stride) {
        const int anchor = i / 20;                   // C/4 = 20 chunks per anchor
        const int c0     = (i - anchor * 20) * 4;
        const int t      = tp[anchor];
        if (t == -2) continue;                       // ignore band: zero loss
        const f4v p4 = __builtin_nontemporal_load(cp + i);
        float pv[4] = {p4.x, p4.y, p4.z, p4.w};
        #pragma unroll
        for (int j = 0; j < 4; ++j) {
            const float p = fminf(fmaxf(pv[j], 1e-4f), 1.0f - 1e-4f);
            if (c0 + j == t) {                       // target == 1
                const float q = 1.0f - p;
                sum += ALPHA_ * q * q * (-__logf(p));
            } else {                                 // target == 0
                sum += (1.0f - ALPHA_) * p * p * (-__logf(1.0f - p));
            }
        }
    }

    const float bs = block_reduce_sum(sum, s_red);
    if (threadIdx.x == 0) atomicAdd(&acc[b], bs);
}

// ---------------- Kernel C: finalize ----------------
__global__ void fl_finalize_kernel(const float* __restrict__ acc,
                                   const float* __restrict__ annotations,
                                   float* __restrict__ out)
{
    if (threadIdx.x == 0 && blockIdx.x == 0) {
        float cm = 0.0f, rm = 0.0f;
        for (int b = 0; b < Bc; ++b) {
            bool hv = false;
            for (int j = 0; j < Mc; ++j)
                if (annotations[((size_t)b * Mc + j) * 5 + 4] != -1.0f) hv = true;
            const float np  = acc[2 * Bc + b];
            const float den = fmaxf(np, 1.0f);
            const float clt = hv ? acc[b] / den : 0.0f;
            const float rgt = (np > 0.0f) ? acc[Bc + b] / (4.0f * den) : 0.0f;
            cm += clt;
            rm += rgt;
        }
        out[0] = cm / (float)Bc;
        out[1] = rm / (float)Bc;
    }
}

// ---------------- launcher ----------------
extern "C" void kernel_launch(void* const* d_in, const int* in_sizes, int n_in,
                              void* d_out, int out_size, void* d_ws, size_t ws_size,
                              hipStream_t stream) {
    const float* cls     = (const float*)d_in[0];   // [B,A,C]
    const float* regs    = (const float*)d_in[1];   // [B,A,4]
    const float* anchors = (const float*)d_in[2];   // [1,A,4]
    const float* ann     = (const float*)d_in[3];   // [B,M,5]
    float* out = (float*)d_out;

    float* acc = (float*)d_ws;                          // 24 accumulators
    signed char* targ = (signed char*)d_ws + 256;       // [B,A] target codes

    (void)hipMemsetAsync(acc, 0, 256, stream);

    fl_assign_kernel<<<dim3((Ac + 255) / 256, Bc), 256, 0, stream>>>(
        anchors, ann, regs, targ, acc);

    fl_cls_kernel<<<dim3(2048, Bc), 256, 0, stream>>>(cls, targ, acc);

    fl_finalize_kernel<<<1, 32, 0, stream>>>(acc, ann, out);
}